// Atten2Map_57062935494715
// MI455X (gfx1250) — compile-verified
//
#include <hip/hip_runtime.h>
#include <hip/hip_bf16.h>
#include <math.h>

// ---- problem constants (from reference) ----
#define NNEI   60
#define NI     128
#define NCOL   256     // ND * 2 * NH = 32 * 8
#define NH     4
#define SHIFT  20.0f

// LDS strides chosen to avoid bank conflicts (64 banks of 4B):
//  G2  stride 132 -> row step mod 64 = 4 (conflict-free 16-lane column reads)
//  QK  stride 260 -> row step mod 64 = 4 (conflict-free strided head reads)
#define G2_STRIDE 132
#define QK_STRIDE 260
#define SMEM_FLOATS (64*G2_STRIDE + 64*QK_STRIDE + 64*3 + 64 + 64)

typedef __attribute__((ext_vector_type(2))) float v2f;
typedef __attribute__((ext_vector_type(8))) float v8f;

__global__ void __launch_bounds__(256)
atten2map_kernel(const float* __restrict__ g2,
                 const float* __restrict__ h2,
                 const unsigned char* __restrict__ nmask,
                 const float* __restrict__ sw,
                 const float* __restrict__ Wqk,
                 float* __restrict__ out)
{
    const int bl   = blockIdx.x;           // (b*nloc + l), 0..4095
    const int tid  = threadIdx.x;
    const int wave = tid >> 5;             // 0..7 (wave32)
    const int lane = tid & 31;
    const int half = lane >> 4;            // 0: lanes 0-15, 1: lanes 16-31
    const int l16  = lane & 15;

    extern __shared__ float smem[];
    float* G2  = smem;                     // 64 x G2_STRIDE
    float* QK  = G2  + 64 * G2_STRIDE;     // 64 x QK_STRIDE (g2qk, 64x256 used)
    float* H2S = QK  + 64 * QK_STRIDE;     // 64 x 3 (rows >=60 zero)
    float* SWS = H2S + 64 * 3;             // 64      (>=60 zero)
    float* MSK = SWS + 64;                 // 64      (0/1 floats, >=60 zero)

    // ---------------- stage inputs into LDS ----------------
    const long ibase = (long)bl * NNEI * NI;
    for (int idx = tid; idx < 64 * NI; idx += 256) {
        const int r = idx >> 7, c = idx & 127;
        G2[r * G2_STRIDE + c] = (r < NNEI) ? g2[ibase + (long)r * NI + c] : 0.0f;
    }
    for (int idx = tid; idx < 64 * 3; idx += 256) {
        H2S[idx] = (idx < NNEI * 3) ? h2[(long)bl * (NNEI * 3) + idx] : 0.0f;
    }
    if (tid < 64) {
        SWS[tid] = (tid < NNEI) ? sw[(long)bl * NNEI + tid] : 0.0f;
        MSK[tid] = (tid < NNEI) ? (nmask[(long)bl * NNEI + tid] ? 1.0f : 0.0f) : 0.0f;
    }
    __syncthreads();

    // ---------------- GEMM1: g2qk(64x256) = G2(64x128) x Wqk(128x256) ------
    // wave w owns output columns [32w, 32w+32): 2 N-tiles x 4 M-tiles.
    const int ncol0 = wave * 32;
    v8f acc1[2][4];
#pragma unroll
    for (int nn = 0; nn < 2; ++nn)
#pragma unroll
        for (int m = 0; m < 4; ++m) acc1[nn][m] = (v8f)0.0f;

    for (int kk = 0; kk < NI / 4; ++kk) {          // 32 K-steps of 4
        const int krow = kk * 4 + half * 2;        // K rows for v0/v1 per half
        v2f a[4];
#pragma unroll
        for (int m = 0; m < 4; ++m) {              // A frag: 16x4 from LDS
            const float* p = &G2[(m * 16 + l16) * G2_STRIDE + krow];
            a[m][0] = p[0];
            a[m][1] = p[1];
        }
#pragma unroll
        for (int nn = 0; nn < 2; ++nn) {           // B frag: 4x16 from global
            const int c = ncol0 + nn * 16 + l16;
            v2f b;
            b[0] = Wqk[(long)krow * NCOL + c];
            b[1] = Wqk[(long)(krow + 1) * NCOL + c];
#pragma unroll
            for (int m = 0; m < 4; ++m) {
                acc1[nn][m] = __builtin_amdgcn_wmma_f32_16x16x4_f32(
                    false, a[m], false, b, (short)0, acc1[nn][m], false, false);
            }
        }
    }

    // spill g2qk to LDS (C/D layout: row = m*16 + half*8 + v, col striped l16)
#pragma unroll
    for (int nn = 0; nn < 2; ++nn)
#pragma unroll
        for (int m = 0; m < 4; ++m)
#pragma unroll
            for (int v = 0; v < 8; ++v) {
                const int r = m * 16 + half * 8 + v;
                const int c = ncol0 + nn * 16 + l16;
                QK[r * QK_STRIDE + c] = acc1[nn][m][v];
            }
    __syncthreads();

    // ---------------- GEMM2: per head, attn(64x64) = Q(64x32) x K^T --------
    // wave w -> head h = w>>1, M-half mbase = (w&1)*2; wave owns full rows.
    const int h     = wave >> 1;
    const int mbase = (wave & 1) * 2;
    v8f acc2[2][4];
#pragma unroll
    for (int mi = 0; mi < 2; ++mi)
#pragma unroll
        for (int n = 0; n < 4; ++n) acc2[mi][n] = (v8f)0.0f;

    for (int kk = 0; kk < 8; ++kk) {               // K = ND = 32, 4 per step
        const int d0 = kk * 4 + half * 2;
        v2f a[2], bf[4];
#pragma unroll
        for (int mi = 0; mi < 2; ++mi) {           // Q frag: q[r][d] = QK[r][d*8+h]
            const int r = (mbase + mi) * 16 + l16;
            a[mi][0] = QK[r * QK_STRIDE + d0 * 8 + h];
            a[mi][1] = QK[r * QK_STRIDE + (d0 + 1) * 8 + h];
        }
#pragma unroll
        for (int n = 0; n < 4; ++n) {              // K^T frag: b[d][j] = QK[j][d*8+h+4]
            const int c = n * 16 + l16;
            bf[n][0] = QK[c * QK_STRIDE + d0 * 8 + h + 4];
            bf[n][1] = QK[c * QK_STRIDE + (d0 + 1) * 8 + h + 4];
        }
#pragma unroll
        for (int mi = 0; mi < 2; ++mi)
#pragma unroll
            for (int n = 0; n < 4; ++n) {
                acc2[mi][n] = __builtin_amdgcn_wmma_f32_16x16x4_f32(
                    false, a[mi], false, bf[n], (short)0, acc2[mi][n], false, false);
            }
    }

    // ---------------- fused epilogue: scale, softmax, mask, store ----------
    const float RS32 = 0.17677669529663687f;   // 1/sqrt(32)
    const float RS3  = 0.57735026918962576f;   // 1/sqrt(3)

#pragma unroll
    for (int mi = 0; mi < 2; ++mi) {
#pragma unroll
        for (int v = 0; v < 8; ++v) {
            const int q = (mbase + mi) * 16 + half * 8 + v;   // row index
            const float swq = SWS[q];
            const float hx = H2S[q * 3 + 0], hy = H2S[q * 3 + 1], hz = H2S[q * 3 + 2];

            float t[4], hh[4], swk[4];
            float rmax = -INFINITY;
#pragma unroll
            for (int n = 0; n < 4; ++n) {
                const int j = n * 16 + l16;
                hh[n]  = hx * H2S[j * 3 + 0] + hy * H2S[j * 3 + 1] + hz * H2S[j * 3 + 2];
                swk[n] = SWS[j];
                const float val = acc2[mi][n][v] * RS32 * hh[n];
                t[n] = (val + SHIFT) * swq * swk[n] - SHIFT;
                if (j < NNEI) rmax = fmaxf(rmax, t[n]);
            }
            // row reduction across the 16 lanes sharing this row (wave32 halves)
#pragma unroll
            for (int off = 1; off < 16; off <<= 1)
                rmax = fmaxf(rmax, __shfl_xor(rmax, off, 32));

            float e[4], rsum = 0.0f;
#pragma unroll
            for (int n = 0; n < 4; ++n) {
                const int j = n * 16 + l16;
                e[n] = (j < NNEI) ? __expf(t[n] - rmax) : 0.0f;
                rsum += e[n];
            }
#pragma unroll
            for (int off = 1; off < 16; off <<= 1)
                rsum += __shfl_xor(rsum, off, 32);
            const float inv = 1.0f / rsum;

            if (q < NNEI) {
                const float mq = MSK[q];
                float* orow = out + ((long)bl * NNEI + q) * (NNEI * NH);
#pragma unroll
                for (int n = 0; n < 4; ++n) {
                    const int j = n * 16 + l16;
                    if (j < NNEI) {
                        const float p = e[n] * inv * mq * MSK[j]
                                      * swq * swk[n] * hh[n] * RS3;
                        orow[j * NH + h] = p;   // layout (b,l,q,k,h)
                    }
                }
            }
        }
    }
}

extern "C" void kernel_launch(void* const* d_in, const int* in_sizes, int n_in,
                              void* d_out, int out_size, void* d_ws, size_t ws_size,
                              hipStream_t stream) {
    const float*         g2    = (const float*)d_in[0];
    const float*         h2    = (const float*)d_in[1];
    const unsigned char* nmask = (const unsigned char*)d_in[2];
    const float*         sw    = (const float*)d_in[3];
    const float*         Wqk   = (const float*)d_in[4];
    float*               out   = (float*)d_out;

    const int nbl = in_sizes[3] / NNEI;   // nb * nloc = 4096
    const size_t smem = (size_t)SMEM_FLOATS * sizeof(float);

    atten2map_kernel<<<dim3(nbl), dim3(256), smem, stream>>>(
        g2, h2, nmask, sw, Wqk, out);
}